// FrequencyRepresentationModule_2D_v0512_28802050687302
// MI455X (gfx1250) — compile-verified
//
#include <hip/hip_runtime.h>
#include <hip/hip_bf16.h>

// ---------------------------------------------------------------------------
// CDNA5 / gfx1250 implementation. Heavy lifting = batched complex GEMMs via
// V_WMMA_F32_16X16X4_F32 (fp32 in/out, matches reference precision).
// GEMM is templated on all strides so each variant gets compile-time
// addressing, vectorized (b64) fragment loads where pairs are contiguous,
// and a 4x-unrolled K loop for load/WMMA overlap.
// ---------------------------------------------------------------------------

typedef float v2f __attribute__((ext_vector_type(2)));
typedef float v8f __attribute__((ext_vector_type(8)));

#define BSZ   64
#define HH    64      // SD0 / DIM0
#define WWID  256     // DIM1
#define NPIX  (HH * WWID)        // 16384
#define G_TOT (BSZ * NPIX)       // 1048576   (b,64,256) plane
#define CONV_TOT (BSZ * 2 * NPIX)// 2097152   (b,2,64,256)
#define OH 257
#define OW 1025

// ---------------------------------------------------------------------------
// Generic batched complex GEMM:
//   C[b][m,n] = sum_k A[m,k] * B[b][k,n]   (complex; A shared across batch)
// A[m,k]    at Ar/Ai[m*AMS + k*AKS]
// B[b][k,n] at Br/Bi[b*bBs + k*BKS + n*BNS]
// C[b][m,n] at Cr/Ci[b*cBs + m*CMS + n*CNS]
// EPI==1: EI transform  s = 1 + log(cr^2+ci^2);  out = (s*cr, -s*ci)
// One wave per 16x16 output tile, 4 f32 WMMAs per K=4 chunk.
//
// Fragment layouts (per ISA tables, wave32):
//   A 16x4 : lanes 0-15 -> K = {k0, k0+1}, lanes 16-31 -> K = {k0+2, k0+3}
//   B 4x16 : symmetric (lane halves split K range, VGPR index steps K)
//   C/D    : vgpr j -> row m0 + 8*half + j, col n0 + lane15
// ---------------------------------------------------------------------------
template <int AMS, int AKS, int BKS, int BNS, int CMS, int CNS, int EPI>
__global__ __launch_bounds__(256)
void gemm_cplx(const float* __restrict__ Ar, const float* __restrict__ Ai,
               const float* __restrict__ Br, const float* __restrict__ Bi,
               float* __restrict__ Cr, float* __restrict__ Ci,
               int M, int N, int K, int nB,
               long bBs, long cBs)
{
    int mt = M >> 4, nt = N >> 4;
    int tiles = nB * mt * nt;
    int wave = (int)((blockIdx.x * blockDim.x + threadIdx.x) >> 5);
    if (wave >= tiles) return;           // whole waves retire; active waves EXEC=all-1s
    int lane = threadIdx.x & 31;
    int half = lane >> 4;                // which 16-lane half
    int l    = lane & 15;

    int b  = wave / (mt * nt);
    int t  = wave - b * (mt * nt);
    int m0 = (t / nt) << 4;
    int n0 = (t - (t / nt) * nt) << 4;

    const float* br = Br + (long)b * bBs;
    const float* bi = Bi + (long)b * bBs;

    v8f accR = {0.f,0.f,0.f,0.f,0.f,0.f,0.f,0.f};
    v8f accI = {0.f,0.f,0.f,0.f,0.f,0.f,0.f,0.f};

    const long aRow = (long)(m0 + l) * AMS;
    const long bCol = (long)(n0 + l) * BNS;

    #pragma unroll 4
    for (int k0 = 0; k0 < K; k0 += 4) {
        const int ka = k0 + 2 * half;
        v2f afr, afi, nafi, bfr, bfi;
        if (AKS == 1) {                  // contiguous pair -> single b64 load
            afr = *(const v2f*)(Ar + aRow + ka);
            afi = *(const v2f*)(Ai + aRow + ka);
        } else {
            afr[0] = Ar[aRow + (long)ka * AKS];
            afr[1] = Ar[aRow + (long)(ka + 1) * AKS];
            afi[0] = Ai[aRow + (long)ka * AKS];
            afi[1] = Ai[aRow + (long)(ka + 1) * AKS];
        }
        if (BKS == 1) {                  // contiguous pair -> single b64 load
            bfr = *(const v2f*)(br + bCol + ka);
            bfi = *(const v2f*)(bi + bCol + ka);
        } else {
            bfr[0] = br[(long)ka * BKS + bCol];
            bfr[1] = br[(long)(ka + 1) * BKS + bCol];
            bfi[0] = bi[(long)ka * BKS + bCol];
            bfi[1] = bi[(long)(ka + 1) * BKS + bCol];
        }
        nafi[0] = -afi[0];
        nafi[1] = -afi[1];
        // Cr += Ar*Br ; Cr += (-Ai)*Bi ; Ci += Ar*Bi ; Ci += Ai*Br
        accR = __builtin_amdgcn_wmma_f32_16x16x4_f32(false, afr,  false, bfr, (short)0, accR, false, false);
        accR = __builtin_amdgcn_wmma_f32_16x16x4_f32(false, nafi, false, bfi, (short)0, accR, false, false);
        accI = __builtin_amdgcn_wmma_f32_16x16x4_f32(false, afr,  false, bfi, (short)0, accI, false, false);
        accI = __builtin_amdgcn_wmma_f32_16x16x4_f32(false, afi,  false, bfr, (short)0, accI, false, false);
    }

    float* cr = Cr + (long)b * cBs;
    float* ci = Ci + (long)b * cBs;
    #pragma unroll
    for (int j = 0; j < 8; ++j) {
        int m = m0 + 8 * half + j;
        long off = (long)m * CMS + (long)(n0 + l) * CNS;
        float vr = accR[j], vi = accI[j];
        if (EPI == 1) {
            float s = 1.0f + logf(vr * vr + vi * vi);
            vr = s * vr;
            vi = -s * vi;
        }
        cr[off] = vr;
        ci[off] = vi;
    }
}

// ---------------------------------------------------------------------------
// Fold W_pre into W_in:  Weff (256 x 64) complex
// ---------------------------------------------------------------------------
__global__ void build_weff(const float* __restrict__ Winr, const float* __restrict__ Wini,
                           const float* __restrict__ Wpr,  const float* __restrict__ Wpi,
                           float* __restrict__ Weffr, float* __restrict__ Weffi)
{
    int idx = blockIdx.x * blockDim.x + threadIdx.x;   // 256*64
    if (idx >= 256 * 64) return;
    int wp = idx >> 6, w = idx & 63;
    float sr = 0.f, si = 0.f;
    #pragma unroll
    for (int a = 0; a < 2; ++a) {
        float wr = Winr[(a * 256 + wp) * 64 + w];
        float wi = Wini[(a * 256 + wp) * 64 + w];
        sr += Wpr[a] * wr - Wpi[a] * wi;
        si += Wpr[a] * wi + Wpi[a] * wr;
    }
    Weffr[idx] = sr;
    Weffi[idx] = si;
}

// ---------------------------------------------------------------------------
// Per (b,h): w-reduction of G0 * z, atan2, rotate original G0 -> branch out
// One block per b; 256 threads == 256 w columns.
// ---------------------------------------------------------------------------
__global__ __launch_bounds__(256)
void phase_reduce_rotate(const float* __restrict__ G0r, const float* __restrict__ G0i,
                         const float* __restrict__ Zr,  const float* __restrict__ Zi,
                         float* __restrict__ Xbr, float* __restrict__ Xbi)
{
    int b = blockIdx.x;
    int t = threadIdx.x;
    __shared__ float s0[256], s1[256];
    __shared__ float cc[2];
    const float* g0r = G0r + (long)b * NPIX;
    const float* g0i = G0i + (long)b * NPIX;
    const float* zrb = Zr  + (long)b * NPIX;
    const float* zib = Zi  + (long)b * NPIX;
    float* xr = Xbr + (long)b * NPIX;
    float* xi = Xbi + (long)b * NPIX;

    for (int h = 0; h < HH; ++h) {
        int off = h * WWID + t;
        float gr = g0r[off], gi = g0i[off];
        float zr = zrb[off], zi = zib[off];
        s0[t] = gr * zr - gi * zi;
        s1[t] = gr * zi + gi * zr;
        __syncthreads();
        for (int s = 128; s > 0; s >>= 1) {
            if (t < s) { s0[t] += s0[t + s]; s1[t] += s1[t + s]; }
            __syncthreads();
        }
        if (t == 0) {
            float ang = atan2f(s1[0], s0[0]);
            cc[0] = cosf(ang);
            cc[1] = -sinf(ang);
        }
        __syncthreads();
        float cr = cc[0], ci = cc[1];
        xr[off] = gr * cr - gi * ci;
        xi[off] = gr * ci + gi * cr;
        __syncthreads();
    }
}

// ---------------------------------------------------------------------------
// x = Wc[0]*x1 + Wc[1]*x2  (complex, per layer)
// ---------------------------------------------------------------------------
__global__ void combine_branches(const float* __restrict__ X1r, const float* __restrict__ X1i,
                                 const float* __restrict__ X2r, const float* __restrict__ X2i,
                                 const float* __restrict__ Wcr, const float* __restrict__ Wci,
                                 int layer,
                                 float* __restrict__ Xr, float* __restrict__ Xi)
{
    int idx = blockIdx.x * blockDim.x + threadIdx.x;
    if (idx >= G_TOT) return;
    float c0r = Wcr[layer * 2 + 0], c1r = Wcr[layer * 2 + 1];
    float c0i = Wci[layer * 2 + 0], c1i = Wci[layer * 2 + 1];
    float a0r = X1r[idx], a0i = X1i[idx], a1r = X2r[idx], a1i = X2i[idx];
    Xr[idx] = c0r * a0r + c1r * a1r - c0i * a0i - c1i * a1i;
    Xi[idx] = c0r * a0i + c1r * a1i + c0i * a0r + c1i * a1r;
}

// ---------------------------------------------------------------------------
// Per-b max-normalized magnitude: x = sqrt(fr^2+fi^2) / max_b(|f|)
// ---------------------------------------------------------------------------
__global__ __launch_bounds__(256)
void norm_mag(const float* __restrict__ Fr, const float* __restrict__ Fi, float* __restrict__ X)
{
    int b = blockIdx.x;
    int t = threadIdx.x;
    __shared__ float sm[256];
    const float* fr = Fr + (long)b * 32768;
    const float* fi = Fi + (long)b * 32768;
    float* x = X + (long)b * 32768;
    float mx = 0.f;
    for (int i = t; i < 32768; i += 256) {
        float r = fr[i], im = fi[i];
        float mag = sqrtf(r * r + im * im);
        x[i] = mag;
        mx = fmaxf(mx, mag);
    }
    sm[t] = mx;
    __syncthreads();
    for (int s = 128; s > 0; s >>= 1) {
        if (t < s) sm[t] = fmaxf(sm[t], sm[t + s]);
        __syncthreads();
    }
    float m = sm[0];
    for (int i = t; i < 32768; i += 256) x[i] /= m;
}

// ---------------------------------------------------------------------------
// 3x3 circular conv, 2->2 channels, (64,2,64,256)
// ---------------------------------------------------------------------------
__global__ void conv3x3_circ(const float* __restrict__ in, const float* __restrict__ w,
                             float* __restrict__ out)
{
    int idx = blockIdx.x * blockDim.x + threadIdx.x;
    if (idx >= CONV_TOT) return;
    int x = idx & 255;
    int y = (idx >> 8) & 63;
    int o = (idx >> 14) & 1;
    int b = idx >> 15;
    const float* base = in + (long)b * (2 * NPIX);
    float acc = 0.f;
    #pragma unroll
    for (int i = 0; i < 2; ++i) {
        const float* wp = w + (o * 2 + i) * 9;
        const float* ip = base + i * NPIX;
        #pragma unroll
        for (int dy = 0; dy < 3; ++dy) {
            int yy = (y + dy + 63) & 63;
            #pragma unroll
            for (int dx = 0; dx < 3; ++dx) {
                int xx = (x + dx + 255) & 255;
                acc += ip[yy * 256 + xx] * wp[dy * 3 + dx];
            }
        }
    }
    out[idx] = acc;
}

// ---------------------------------------------------------------------------
// Batch-norm statistics over (b,h,w) per channel; one block per channel.
// stats[ch*2] = mu, stats[ch*2+1] = rsqrt(var+eps)
// ---------------------------------------------------------------------------
__global__ __launch_bounds__(256)
void bn_stats(const float* __restrict__ h, float* __restrict__ stats)
{
    int ch = blockIdx.x;
    int t = threadIdx.x;
    __shared__ float ss[256], sq[256];
    float s = 0.f, q = 0.f;
    for (int b = 0; b < BSZ; ++b) {
        const float* p = h + (long)b * (2 * NPIX) + (long)ch * NPIX;
        for (int i = t; i < NPIX; i += 256) {
            float v = p[i];
            s += v;
            q += v * v;
        }
    }
    ss[t] = s; sq[t] = q;
    __syncthreads();
    for (int st = 128; st > 0; st >>= 1) {
        if (t < st) { ss[t] += ss[t + st]; sq[t] += sq[t + st]; }
        __syncthreads();
    }
    if (t == 0) {
        float N = (float)(BSZ * NPIX);
        float mu = ss[0] / N;
        float var = sq[0] / N - mu * mu;
        stats[ch * 2 + 0] = mu;
        stats[ch * 2 + 1] = rsqrtf(var + 1e-5f);
    }
}

__global__ void bn_apply(const float* __restrict__ h, const float* __restrict__ stats,
                         const float* __restrict__ g, const float* __restrict__ bta,
                         float* __restrict__ out, int do_relu)
{
    int idx = blockIdx.x * blockDim.x + threadIdx.x;
    if (idx >= CONV_TOT) return;
    int ch = (idx >> 14) & 1;
    float v = (h[idx] - stats[ch * 2]) * stats[ch * 2 + 1] * g[ch] + bta[ch];
    out[idx] = do_relu ? fmaxf(v, 0.f) : v;
}

__global__ void bn_residual_relu(const float* __restrict__ h, const float* __restrict__ stats,
                                 const float* __restrict__ g, const float* __restrict__ bta,
                                 float* __restrict__ x)
{
    int idx = blockIdx.x * blockDim.x + threadIdx.x;
    if (idx >= CONV_TOT) return;
    int ch = (idx >> 14) & 1;
    float v = (h[idx] - stats[ch * 2]) * stats[ch * 2 + 1] * g[ch] + bta[ch];
    x[idx] = fmaxf(x[idx] + v, 0.f);
}

// ---------------------------------------------------------------------------
// Transposed conv: lhs_dilation=4, kernel 8x8, pad (5,6), 2ch -> 1ch
// out: (64, 257, 1025)
// ---------------------------------------------------------------------------
__global__ void upsample_tconv(const float* __restrict__ x, const float* __restrict__ Wout,
                               float* __restrict__ out)
{
    long idx = (long)blockIdx.x * blockDim.x + threadIdx.x;
    long total = (long)BSZ * OH * OW;
    if (idx >= total) return;
    int ox = (int)(idx % OW);
    int oy = (int)((idx / OW) % OH);
    int b  = (int)(idx / ((long)OH * OW));
    const float* xb = x + (long)b * (2 * NPIX);
    int ty = oy - 5, tx = ox - 5;
    float acc = 0.f;
    for (int ky = 0; ky < 8; ++ky) {
        int py = ty + ky;
        if (py < 0 || py > 252 || (py & 3)) continue;
        int iy = py >> 2;
        for (int kx = 0; kx < 8; ++kx) {
            int px = tx + kx;
            if (px < 0 || px > 1020 || (px & 3)) continue;
            int ix = px >> 2;
            #pragma unroll
            for (int c = 0; c < 2; ++c)
                acc += xb[c * NPIX + iy * 256 + ix] * Wout[c * 64 + (7 - ky) * 8 + (7 - kx)];
        }
    }
    out[idx] = acc;
}

// ---------------------------------------------------------------------------
// Host side
// ---------------------------------------------------------------------------
template <int AMS, int AKS, int BKS, int BNS, int CMS, int CNS, int EPI>
static inline void launch_gemm(const float* Ar, const float* Ai,
                               const float* Br, const float* Bi,
                               float* Cr, float* Ci,
                               int M, int N, int K, int nB,
                               long bBs, long cBs, hipStream_t stream)
{
    int tiles = nB * (M / 16) * (N / 16);
    int blocks = (tiles + 7) / 8;   // 8 waves (wave32) per 256-thread block
    gemm_cplx<AMS, AKS, BKS, BNS, CMS, CNS, EPI>
        <<<blocks, 256, 0, stream>>>(Ar, Ai, Br, Bi, Cr, Ci, M, N, K, nB, bBs, cBs);
}

extern "C" void kernel_launch(void* const* d_in, const int* in_sizes, int n_in,
                              void* d_out, int out_size, void* d_ws, size_t ws_size,
                              hipStream_t stream)
{
    const float* inp     = (const float*)d_in[0];
    const float* W_in_r  = (const float*)d_in[1];
    const float* W_in_i  = (const float*)d_in[2];
    const float* W_pre_r = (const float*)d_in[3];
    const float* W_pre_i = (const float*)d_in[4];
    const float* Wm_r    = (const float*)d_in[5];
    const float* Wm_i    = (const float*)d_in[6];
    const float* Wc_r    = (const float*)d_in[7];
    const float* Wc_i    = (const float*)d_in[8];
    const float* W_l3_r  = (const float*)d_in[9];
    const float* W_l3_i  = (const float*)d_in[10];
    const float* conv1_w = (const float*)d_in[11];
    const float* bn1_g   = (const float*)d_in[12];
    const float* bn1_b   = (const float*)d_in[13];
    const float* conv2_w = (const float*)d_in[14];
    const float* bn2_g   = (const float*)d_in[15];
    const float* bn2_b   = (const float*)d_in[16];
    const float* W_out   = (const float*)d_in[17];
    float* ws  = (float*)d_ws;
    float* out = (float*)d_out;

    // workspace layout (float offsets)
    const size_t WEFFR = 0;
    const size_t WEFFI = WEFFR + 256 * 64;
    const size_t G0R   = WEFFI + 256 * 64;
    const size_t G0I   = G0R + G_TOT;
    const size_t XR    = G0I + G_TOT;     // current x_real
    const size_t XI    = XR + G_TOT;
    const size_t EIR   = XI + G_TOT;
    const size_t EII   = EIR + G_TOT;
    const size_t ZR    = EII + G_TOT;
    const size_t ZI    = ZR + G_TOT;
    const size_t X1R   = ZI + G_TOT;
    const size_t X1I   = X1R + G_TOT;
    const size_t X2R   = X1I + G_TOT;
    const size_t X2I   = X2R + G_TOT;
    const size_t FR    = X2I + G_TOT;     // (b,128,256); reused as conv H1
    const size_t FI    = FR + (size_t)BSZ * 128 * 256;   // reused as conv H2
    const size_t XC    = FI + (size_t)BSZ * 128 * 256;   // (b,2,64,256)
    const size_t STATS = XC + CONV_TOT;

    // ---- Stage 1: fold W_pre into W_in, then one batched complex GEMM ----
    build_weff<<<64, 256, 0, stream>>>(W_in_r, W_in_i, W_pre_r, W_pre_i, ws + WEFFR, ws + WEFFI);

    // C[b][m=w'(256), n=h(64)] = sum_w Weff[w',w] * X[b][h,w]   -> G0[b,h,w']
    launch_gemm<64, 1, 1, 64, 1, 256, 0>(
        ws + WEFFR, ws + WEFFI,
        inp, inp + 4096,                      // xr / xi planes within each batch
        ws + G0R, ws + G0I,
        /*M=*/256, /*N=*/64, /*K=*/64, /*nB=*/BSZ,
        /*bBs=*/8192, /*cBs=*/NPIX, stream);

    // x_cur = G0
    hipMemcpyAsync(ws + XR, ws + G0R, sizeof(float) * G_TOT, hipMemcpyDeviceToDevice, stream);
    hipMemcpyAsync(ws + XI, ws + G0I, sizeof(float) * G_TOT, hipMemcpyDeviceToDevice, stream);

    // ---- Phase layers ----
    const int gridPW = (G_TOT + 255) / 256;
    for (int layer = 0; layer < 4; ++layer) {
        for (int br = 0; br < 2; ++br) {
            int wsel = layer + br * 4;
            const float* wr = Wm_r + (size_t)wsel * 64 * 64;
            const float* wi = Wm_i + (size_t)wsel * 64 * 64;
            // I = W x X_cur  (complex), EI epilogue: s=1+log|I|^2, (s*Ir, -s*Ii)
            launch_gemm<64, 1, 256, 1, 256, 1, 1>(
                wr, wi, ws + XR, ws + XI, ws + EIR, ws + EII,
                64, 256, 64, BSZ, NPIX, NPIX, stream);
            // z = W^T x EI   (A[m=h,k=c] = W[c,h] -> swapped strides)
            launch_gemm<1, 64, 256, 1, 256, 1, 0>(
                wr, wi, ws + EIR, ws + EII, ws + ZR, ws + ZI,
                64, 256, 64, BSZ, NPIX, NPIX, stream);
            // reduce over w, atan2, rotate original G0 -> branch output
            float* xbr = ws + (br == 0 ? X1R : X2R);
            float* xbi = ws + (br == 0 ? X1I : X2I);
            phase_reduce_rotate<<<BSZ, 256, 0, stream>>>(ws + G0R, ws + G0I,
                                                         ws + ZR, ws + ZI, xbr, xbi);
        }
        combine_branches<<<gridPW, 256, 0, stream>>>(ws + X1R, ws + X1I, ws + X2R, ws + X2I,
                                                     Wc_r, Wc_i, layer, ws + XR, ws + XI);
    }

    // ---- Final projection (128x64) x (64x256) complex ----
    launch_gemm<64, 1, 256, 1, 256, 1, 0>(
        W_l3_r, W_l3_i, ws + XR, ws + XI, ws + FR, ws + FI,
        128, 256, 64, BSZ, NPIX, (long)32768, stream);

    // magnitude + per-batch max normalization -> (b,2,64,256)
    norm_mag<<<BSZ, 256, 0, stream>>>(ws + FR, ws + FI, ws + XC);

    // ---- Residual conv blocks (F buffers reused as scratch) ----
    float* H1 = ws + FR;
    float* H2 = ws + FI;
    const int gridCV = (CONV_TOT + 255) / 256;
    for (int layer = 0; layer < 4; ++layer) {
        conv3x3_circ<<<gridCV, 256, 0, stream>>>(ws + XC, conv1_w + (size_t)layer * 36, H1);
        bn_stats<<<2, 256, 0, stream>>>(H1, ws + STATS);
        bn_apply<<<gridCV, 256, 0, stream>>>(H1, ws + STATS, bn1_g + layer * 2, bn1_b + layer * 2,
                                             H2, /*relu=*/1);
        conv3x3_circ<<<gridCV, 256, 0, stream>>>(H2, conv2_w + (size_t)layer * 36, H1);
        bn_stats<<<2, 256, 0, stream>>>(H1, ws + STATS);
        bn_residual_relu<<<gridCV, 256, 0, stream>>>(H1, ws + STATS, bn2_g + layer * 2,
                                                     bn2_b + layer * 2, ws + XC);
    }

    // ---- Transposed-conv upsample -> (64, 257, 1025) ----
    long totalOut = (long)BSZ * OH * OW;
    int gridUP = (int)((totalOut + 255) / 256);
    upsample_tconv<<<gridUP, 256, 0, stream>>>(ws + XC, W_out, out);
}